// ChunkwiseRetention_44495861186728
// MI455X (gfx1250) — compile-verified
//
#include <hip/hip_runtime.h>
#include <hip/hip_bf16.h>

#define BATCH 8
#define SEQ   4000
#define FEAT  256
#define DIM   256
#define BCH   5
#define NCH   800          // SEQ / BCH
#define GAMMA 0.9865f
#define EB    32           // e-columns per scan block
#define GRP   100          // chunks per scan group
#define NGRP  8            // NCH / GRP
#define QSTR  260          // padded Q row stride (floats) -> conflict-free A-frag loads

typedef float v2f __attribute__((ext_vector_type(2)));
typedef float v8f __attribute__((ext_vector_type(8)));

// ---------------------------------------------------------------------------
// CDNA5 async global->LDS copy (ASYNCcnt path, §15.18.3 op 98 / async doc §4).
// ---------------------------------------------------------------------------
__device__ __forceinline__ void async_g2l_b128(unsigned lds_addr, const void* gsrc) {
    unsigned long long ga = (unsigned long long)gsrc;
    asm volatile("global_load_async_to_lds_b128 %0, %1, off"
                 :: "v"(lds_addr), "v"(ga) : "memory");
}
__device__ __forceinline__ void wait_async0() {
    asm volatile("s_wait_asynccnt 0x0" ::: "memory");
}
__device__ __forceinline__ unsigned lds_addr_of(const void* p) {
    return (unsigned)(unsigned long long)p;   // LDS offset lives in the low 32 bits
}

// ---------------------------------------------------------------------------
// Kernel A: fused Q/K/V projection GEMM, full-precision f32 WMMA 16x16x4.
// ---------------------------------------------------------------------------
__global__ __launch_bounds__(256)
void proj_kernel(const float* __restrict__ xq, const float* __restrict__ xk,
                 const float* __restrict__ xv,
                 const float* __restrict__ Wq, const float* __restrict__ Wk,
                 const float* __restrict__ Wv,
                 float* __restrict__ Qo, float* __restrict__ Ko,
                 float* __restrict__ Vo)
{
    const float* X; const float* W; float* O;
    if (blockIdx.z == 0)      { X = xq; W = Wq; O = Qo; }
    else if (blockIdx.z == 1) { X = xk; W = Wk; O = Ko; }
    else                      { X = xv; W = Wv; O = Vo; }

    const int lane = threadIdx.x & 31;
    const int wave = threadIdx.x >> 5;
    const int m0   = blockIdx.x * 16;
    const int n0   = blockIdx.y * 128 + wave * 16;
    const int rc   = lane & 15;
    const int half = lane >> 4;
    const int kofs = half * 2;

    const float* __restrict__ Arow = X + (size_t)(m0 + rc) * FEAT;
    const float* __restrict__ Bcol = W + (n0 + rc);

    v8f acc = {};
    for (int k = 0; k < FEAT; k += 4) {
        v2f a, b;
        a.x = Arow[k + kofs];
        a.y = Arow[k + kofs + 1];
        b.x = Bcol[(size_t)(k + kofs) * DIM];
        b.y = Bcol[(size_t)(k + kofs + 1) * DIM];
        acc = __builtin_amdgcn_wmma_f32_16x16x4_f32(
            false, a, false, b, (short)0, acc, false, false);
    }

    float* __restrict__ Orow = O + (size_t)(m0 + half * 8) * DIM + n0 + rc;
    #pragma unroll
    for (int r = 0; r < 8; ++r)
        Orow[(size_t)r * DIM] = acc[r];
}

// ---------------------------------------------------------------------------
// Kernel B: intra-chunk term  out = (Q K^T ∘ L) V   (writes every element)
// ---------------------------------------------------------------------------
__global__ __launch_bounds__(256)
void intra_kernel(const float* __restrict__ Q, const float* __restrict__ K,
                  const float* __restrict__ V, float* __restrict__ out)
{
    __shared__ float part[25][8];
    __shared__ float SL[5][5];
    const int c  = blockIdx.x;
    const int b  = c / NCH;
    const int ch = c % NCH;
    const int rowbase = b * SEQ + ch * BCH;
    const int t = threadIdx.x;

    const int p   = t >> 3;
    const int sub = t & 7;
    if (p < 25) {
        const int i = p / 5, j = p % 5;
        const float* __restrict__ q = Q + (size_t)(rowbase + i) * DIM;
        const float* __restrict__ k = K + (size_t)(rowbase + j) * DIM;
        float s = 0.f;
        for (int d = sub; d < DIM; d += 8) s += q[d] * k[d];
        part[p][sub] = s;
    }
    __syncthreads();
    if (t < 25) {
        const int i = t / 5, j = t % 5;
        float s = 0.f;
        #pragma unroll
        for (int u = 0; u < 8; ++u) s += part[t][u];
        SL[i][j] = (j <= i) ? s * powf(GAMMA, (float)(j - i)) : 0.f;
    }
    __syncthreads();

    const int d = t;
    float v[5];
    #pragma unroll
    for (int j = 0; j < 5; ++j) v[j] = V[(size_t)(rowbase + j) * DIM + d];
    #pragma unroll
    for (int i = 0; i < 5; ++i) {
        float acc = 0.f;
        #pragma unroll
        for (int j = 0; j < 5; ++j)
            if (j <= i) acc += SL[i][j] * v[j];
        out[(size_t)(rowbase + i) * DIM + d] = acc;
    }
}

// ---------------------------------------------------------------------------
// Phase 1: per-group partial state, double-buffered async K staging.
// grid = (BATCH, NGRP, DIM/EB), block = 256.
// ---------------------------------------------------------------------------
__global__ __launch_bounds__(256)
void scan_phase1(const float* __restrict__ K, const float* __restrict__ V,
                 float* __restrict__ Rloc)
{
    __shared__ __attribute__((aligned(16))) float Rsm[DIM * EB];
    __shared__ __attribute__((aligned(16))) float Ks[2][5 * DIM];
    __shared__ float Vzs[5 * EB];

    const int b  = blockIdx.x;
    const int g  = blockIdx.y;
    const int e0 = blockIdx.z * EB;
    const int t  = threadIdx.x;
    const float gB = powf(GAMMA, (float)BCH);
    const unsigned ks_base = lds_addr_of(Ks);

    auto stageK = [&](int buf, int ii) {
        const float* Kg = K + (size_t)(b * SEQ + ii * BCH) * DIM;
        const unsigned ksb = ks_base + (unsigned)buf * (5 * DIM * 4);
        for (int f = t; f < 5 * DIM / 4; f += 256)
            async_g2l_b128(ksb + f * 16, Kg + f * 4);
    };

    for (int idx = t; idx < DIM * EB; idx += 256) Rsm[idx] = 0.f;
    stageK(0, g * GRP);

    for (int i = g * GRP; i < (g + 1) * GRP; ++i) {
        const int cur = (i - g * GRP) & 1;
        if (t < 5 * EB) {
            const int j = t / EB, e = t % EB;
            const float Z = powf(GAMMA, (float)BCH - (float)i - 1.0f);
            Vzs[j * EB + e] = V[(size_t)(b * SEQ + i * BCH + j) * DIM + e0 + e] * Z;
        }
        wait_async0();
        __syncthreads();
        if (i + 1 < (g + 1) * GRP) stageK(cur ^ 1, i + 1);  // overlaps the update

        for (int idx = t; idx < DIM * EB; idx += 256) {
            const int d = idx >> 5;
            const int e = idx & (EB - 1);
            float r = gB * Rsm[idx];
            r += Ks[cur][0 * DIM + d] * Vzs[0 * EB + e];
            r += Ks[cur][1 * DIM + d] * Vzs[1 * EB + e];
            r += Ks[cur][2 * DIM + d] * Vzs[2 * EB + e];
            r += Ks[cur][3 * DIM + d] * Vzs[3 * EB + e];
            r += Ks[cur][4 * DIM + d] * Vzs[4 * EB + e];
            Rsm[idx] = r;
        }
        __syncthreads();
    }

    float* __restrict__ dst = Rloc + ((size_t)(b * NGRP + g)) * DIM * DIM;
    for (int idx = t; idx < DIM * EB; idx += 256) {
        const int d = idx >> 5;
        const int e = idx & (EB - 1);
        dst[(size_t)d * DIM + e0 + e] = Rsm[idx];
    }
}

// ---------------------------------------------------------------------------
// Phase 2: in-place prefix combine  R[g] = R[g] + gB^GRP * R[g-1]
// ---------------------------------------------------------------------------
__global__ __launch_bounds__(256)
void scan_phase2(float* __restrict__ Rloc)
{
    const int b = blockIdx.x;
    const float gBG = powf(GAMMA, (float)(BCH * GRP));
    const int base_m = blockIdx.y * (DIM * DIM / 16);
    float* __restrict__ Rb = Rloc + (size_t)b * NGRP * DIM * DIM;

    for (int m = base_m + threadIdx.x; m < base_m + DIM * DIM / 16; m += 256) {
        float acc = Rb[m];
        for (int g = 1; g < NGRP; ++g) {
            acc = Rb[(size_t)g * DIM * DIM + m] + gBG * acc;
            Rb[(size_t)g * DIM * DIM + m] = acc;
        }
    }
}

// ---------------------------------------------------------------------------
// Phase 3: per-chunk scan; VALU rank-5 update, WMMA cross term.
//   cross_i = gamma^(i+2) * Qpad_{i+1}[16x256] @ R[256x16]  (waves 0/1, e-tiles)
// grid = (BATCH, NGRP, DIM/EB), block = 256.
// ---------------------------------------------------------------------------
__global__ __launch_bounds__(256)
void scan_phase3(const float* __restrict__ Q, const float* __restrict__ K,
                 const float* __restrict__ V, const float* __restrict__ Rloc,
                 float* __restrict__ out)
{
    __shared__ __attribute__((aligned(16))) float Rsm[DIM * EB];
    __shared__ __attribute__((aligned(16))) float Ks[2][5 * DIM];
    __shared__ __attribute__((aligned(16))) float Qs[2][BCH * QSTR];
    __shared__ float Vzs[5 * EB];

    const int b  = blockIdx.x;
    const int g  = blockIdx.y;
    const int e0 = blockIdx.z * EB;
    const int t  = threadIdx.x;
    const float gB = powf(GAMMA, (float)BCH);
    const unsigned ks_base = lds_addr_of(Ks);
    const unsigned qs_base = lds_addr_of(Qs);

    auto stageKQ = [&](int buf, int ii) {
        const int rowbase = b * SEQ + ii * BCH;
        const float* Kg = K + (size_t)rowbase * DIM;
        const float* Qg = Q + (size_t)(rowbase + BCH) * DIM;
        const bool hn = (ii + 1) < NCH;
        const unsigned ksb = ks_base + (unsigned)buf * (5 * DIM * 4);
        const unsigned qsb = qs_base + (unsigned)buf * (BCH * QSTR * 4);
        for (int f = t; f < 5 * DIM / 4; f += 256) {
            async_g2l_b128(ksb + f * 16, Kg + f * 4);
            if (hn) {
                const int j = f >> 6, c2 = f & 63;       // Q row j -> stride QSTR
                async_g2l_b128(qsb + (unsigned)(j * QSTR + c2 * 4) * 4u, Qg + f * 4);
            }
        }
    };

    // seed R with the prefix state of the previous group
    if (g == 0) {
        for (int idx = t; idx < DIM * EB; idx += 256) Rsm[idx] = 0.f;
    } else {
        const float* __restrict__ src =
            Rloc + ((size_t)(b * NGRP + g - 1)) * DIM * DIM;
        for (int idx = t; idx < DIM * EB; idx += 256) {
            const int d = idx >> 5;
            const int e = idx & (EB - 1);
            Rsm[idx] = src[(size_t)d * DIM + e0 + e];
        }
    }
    stageKQ(0, g * GRP);

    for (int i = g * GRP; i < (g + 1) * GRP; ++i) {
        const int rowbase = b * SEQ + i * BCH;
        const int cur = (i - g * GRP) & 1;

        if (t < 5 * EB) {
            const int j = t / EB, e = t % EB;
            const float Z = powf(GAMMA, (float)BCH - (float)i - 1.0f);
            Vzs[j * EB + e] = V[(size_t)(rowbase + j) * DIM + e0 + e] * Z;
        }
        wait_async0();
        __syncthreads();
        if (i + 1 < (g + 1) * GRP) stageKQ(cur ^ 1, i + 1);  // overlaps compute

        // R = gB*R + K^T Vz
        for (int idx = t; idx < DIM * EB; idx += 256) {
            const int d = idx >> 5;
            const int e = idx & (EB - 1);
            float r = gB * Rsm[idx];
            r += Ks[cur][0 * DIM + d] * Vzs[0 * EB + e];
            r += Ks[cur][1 * DIM + d] * Vzs[1 * EB + e];
            r += Ks[cur][2 * DIM + d] * Vzs[2 * EB + e];
            r += Ks[cur][3 * DIM + d] * Vzs[3 * EB + e];
            r += Ks[cur][4 * DIM + d] * Vzs[4 * EB + e];
            Rsm[idx] = r;
        }
        __syncthreads();

        // cross_i via chained f32 WMMA: waves 0/1 own the two 16-col e-tiles
        const int w = t >> 5;
        if ((i + 1) < NCH && w < 2) {
            const int lane = t & 31;
            const int rc   = lane & 15;       // A row / B-D col
            const int half = lane >> 4;
            const int kp   = half * 2;
            const float rowmask = (rc < BCH) ? 1.f : 0.f;   // zero-pad Q rows 5..15
            const float* Qrow = &Qs[cur][(rc < BCH ? rc : 0) * QSTR];
            v8f cacc = {};
            for (int k = 0; k < DIM; k += 4) {
                v2f a = *(const v2f*)(Qrow + k + kp);
                a.x *= rowmask; a.y *= rowmask;
                v2f bb;
                bb.x = Rsm[(k + kp) * EB + w * 16 + rc];
                bb.y = Rsm[(k + kp + 1) * EB + w * 16 + rc];
                cacc = __builtin_amdgcn_wmma_f32_16x16x4_f32(
                    false, a, false, bb, (short)0, cacc, false, false);
            }
            if (half == 0) {       // valid C rows 0..4 live in lanes 0-15, r<5
                const float sc = powf(GAMMA, (float)i + 2.0f);
                float* __restrict__ orow =
                    out + (size_t)rowbase * DIM + e0 + w * 16 + rc;
                #pragma unroll
                for (int r = 0; r < 5; ++r)
                    orow[(size_t)r * DIM] += sc * cacc[r];
            }
        }
        __syncthreads();
    }
}

// ---------------------------------------------------------------------------
extern "C" void kernel_launch(void* const* d_in, const int* in_sizes, int n_in,
                              void* d_out, int out_size, void* d_ws, size_t ws_size,
                              hipStream_t stream) {
    (void)in_sizes; (void)n_in; (void)out_size; (void)ws_size;
    const float* xq = (const float*)d_in[0];
    const float* xk = (const float*)d_in[1];
    const float* xv = (const float*)d_in[2];
    const float* Wq = (const float*)d_in[3];
    const float* Wk = (const float*)d_in[4];
    const float* Wv = (const float*)d_in[5];
    float* out = (float*)d_out;

    const size_t mat = (size_t)BATCH * SEQ * DIM;
    float* Qw   = (float*)d_ws;
    float* Kw   = Qw + mat;
    float* Vw   = Kw + mat;
    float* Rloc = Vw + mat;                         // BATCH*NGRP*DIM*DIM floats

    dim3 gp((BATCH * SEQ) / 16, DIM / 128, 3);
    proj_kernel<<<gp, 256, 0, stream>>>(xq, xk, xv, Wq, Wk, Wv, Qw, Kw, Vw);

    intra_kernel<<<BATCH * NCH, 256, 0, stream>>>(Qw, Kw, Vw, out);

    dim3 g1(BATCH, NGRP, DIM / EB);
    scan_phase1<<<g1, 256, 0, stream>>>(Kw, Vw, Rloc);

    dim3 g2(BATCH, 16);
    scan_phase2<<<g2, 256, 0, stream>>>(Rloc);

    dim3 g3(BATCH, NGRP, DIM / EB);
    scan_phase3<<<g3, 256, 0, stream>>>(Qw, Kw, Vw, Rloc, out);
}